// IntraAgg_17703855194587
// MI455X (gfx1250) — compile-verified
//
#include <hip/hip_runtime.h>

typedef __attribute__((ext_vector_type(2))) float v2f;
typedef __attribute__((ext_vector_type(8))) float v8f;

#define FEAT   256
#define KDIM   512            // 2*FEAT
#define EMB    128
#define KNEIGH 32
#define TM     16             // batch rows per block
#define PAD    4
#define LDK    (KDIM + PAD)   // padded LDS row stride (floats)

// One-off weight transpose: W[512][128] -> Wt[128][512] so the GEMM's
// B-operand pairs (k, k+1) become contiguous global_load_b64.
__global__ __launch_bounds__(256)
void transpose_w_kernel(const float* __restrict__ W, float* __restrict__ Wt)
{
    const int idx = blockIdx.x * 256 + threadIdx.x;   // 0 .. 65535
    const int k = idx >> 7;                            // 0..511
    const int n = idx & 127;                           // 0..127
    Wt[(size_t)n * KDIM + k] = W[idx];                 // coalesced read
}

template <bool WT>
__global__ __launch_bounds__(256)
void sage_fused_kernel(const int* __restrict__ nodes,
                       const int* __restrict__ neigh,
                       const float* __restrict__ feat,
                       const float* __restrict__ Wmat,   // WT ? Wt[128][512] : W[512][128]
                       float* __restrict__ out)
{
    __shared__ float cat[TM * LDK];   // 16 x 516 floats = 33 KB

    const int tid     = threadIdx.x;
    const int rowBase = blockIdx.x * TM;

    // ---------- Phase A: build cat = [self | mean(neighbors)] in LDS ----------
    // 16 rows x 64 float4-chunks over 256 threads (4 each). Lanes of a wave
    // cover consecutive chunks of the SAME gathered row -> contiguous 512B
    // bursts from the L2-resident (102 MB < 192 MB) feature table.
    #pragma unroll
    for (int p = tid; p < TM * (FEAT / 4); p += 256) {
        const int r  = p >> 6;                 // 0..15 local row
        const int c  = p & 63;                 // float4 chunk 0..63
        const int gr = rowBase + r;

        const float4 sv =
            *((const float4*)(feat + (size_t)nodes[gr] * FEAT) + c);

        float4 acc = make_float4(0.f, 0.f, 0.f, 0.f);
        const int* nrow = neigh + gr * KNEIGH;
        #pragma unroll 4
        for (int t = 0; t < KNEIGH; ++t) {
            const float4 nv =
                *((const float4*)(feat + (size_t)nrow[t] * FEAT) + c);
            acc.x += nv.x; acc.y += nv.y; acc.z += nv.z; acc.w += nv.w;
        }
        const float s = 1.0f / (float)KNEIGH;

        float* dstS = &cat[r * LDK + 4 * c];
        dstS[0] = sv.x; dstS[1] = sv.y; dstS[2] = sv.z; dstS[3] = sv.w;

        float* dstA = &cat[r * LDK + FEAT + 4 * c];
        dstA[0] = acc.x * s; dstA[1] = acc.y * s;
        dstA[2] = acc.z * s; dstA[3] = acc.w * s;
    }
    __syncthreads();

    // ---------- Phase B: fp32 WMMA GEMM, wave w -> output cols [16w,16w+16) ----
    const int wave = tid >> 5;
    const int lane = tid & 31;
    const int m    = lane & 15;              // A: lanes 0-15 and 16-31 both M=0..15
    const int koff = (lane >> 4) << 1;       // lanes 0-15: K=0,1 ; lanes 16-31: K=2,3
    const int n    = wave * 16 + m;          // B column / D column for this lane

    v8f accD = {};
    const float* bbase = WT ? (Wmat + (size_t)n * KDIM)   // Wt row: contiguous in k
                            : (Wmat + n);                 // W column: stride EMB

    #pragma unroll 8
    for (int k0 = 0; k0 < KDIM; k0 += 4) {
        v2f a;
        a.x = cat[m * LDK + k0 + koff];      // contiguous pair -> ds_load_b64
        a.y = cat[m * LDK + k0 + koff + 1];
        v2f b;
        if (WT) {
            const float2 bv = *(const float2*)(bbase + k0 + koff);  // global_load_b64
            b.x = bv.x; b.y = bv.y;
        } else {
            b.x = bbase[(size_t)(k0 + koff) * EMB];
            b.y = bbase[(size_t)(k0 + koff + 1) * EMB];
        }
        // D = A(16x4,f32) x B(4x16,f32) + C  -> v_wmma_f32_16x16x4_f32
        accD = __builtin_amdgcn_wmma_f32_16x16x4_f32(
            false, a, false, b, (short)0, accD, false, false);
    }

    // ---------- Store with ReLU per D layout: VGPR r -> M = r + 8*(lane>=16) ----
    const int mOut = rowBase + ((lane >> 4) << 3);
    float* op = out + (size_t)mOut * EMB + n;
    #pragma unroll
    for (int r = 0; r < 8; ++r) {
        float v = accD[r];
        op[(size_t)r * EMB] = v > 0.f ? v : 0.f;
    }
}

extern "C" void kernel_launch(void* const* d_in, const int* in_sizes, int n_in,
                              void* d_out, int out_size, void* d_ws, size_t ws_size,
                              hipStream_t stream) {
    const int*   nodes = (const int*)d_in[0];
    const int*   neigh = (const int*)d_in[1];
    const float* feat  = (const float*)d_in[2];
    const float* W     = (const float*)d_in[3];
    float*       out   = (float*)d_out;

    const int batch = in_sizes[0];           // 16384
    dim3 grid(batch / TM);                   // 1024 row tiles

    const size_t wtBytes = (size_t)KDIM * EMB * sizeof(float);   // 256 KB
    if (ws_size >= wtBytes) {
        float* Wt = (float*)d_ws;
        transpose_w_kernel<<<dim3((KDIM * EMB) / 256), 256, 0, stream>>>(W, Wt);
        sage_fused_kernel<true><<<grid, 256, 0, stream>>>(nodes, neigh, feat, Wt, out);
    } else {
        sage_fused_kernel<false><<<grid, 256, 0, stream>>>(nodes, neigh, feat, W, out);
    }
}